// SyntaxGuidedAttention_66202625900972
// MI455X (gfx1250) — compile-verified
//
#include <hip/hip_runtime.h>
#include <hip/hip_bf16.h>
#include <math.h>
#include <stdint.h>

typedef float v2f __attribute__((ext_vector_type(2)));
typedef float v8f __attribute__((ext_vector_type(8)));

#define BB 2
#define SS 2048
#define DD 1024
#define HH 16
#define HD 64

// ---------------------------------------------------------------------------
// CDNA5 async copy: global -> LDS, 16 B per lane, tracked by ASYNCcnt.
// Inline asm (portable across ROCm 7.2 / amdgpu-toolchain per bridge doc).
// Generic LDS pointer truncates to the 32-bit LDS offset (ISA 10.2 aperture).
// ---------------------------------------------------------------------------
__device__ __forceinline__ void async_g2l_b128(float* lds_dst, const float* g) {
  const uint32_t l = (uint32_t)(uintptr_t)lds_dst;
  asm volatile("global_load_async_to_lds_b128 %0, %1, off"
               :: "v"(l), "v"(g) : "memory");
}
template <int N>
__device__ __forceinline__ void wait_async() {
  asm volatile("s_wait_asynccnt %0" :: "i"(N) : "memory");
}

// ---------------------------------------------------------------------------
// Tiled GEMM: C[M,N] = A[M,K] @ W[K,N] + bias[N].  M%64==0, N%64==0, K%32==0.
// 256 threads = 8 waves; block tile 64x64; wave = 16x32 (two 16x16 accums).
// K in 32-deep chunks, double-buffered in LDS via async-to-LDS b128.
// A panel rows padded 32->36 floats (16B-aligned, stride !=0 mod 64 banks).
// ---------------------------------------------------------------------------
#define APAD 36
__global__ __launch_bounds__(256)
void wmma_gemm_tiled(const float* __restrict__ A, const float* __restrict__ W,
                     const float* __restrict__ bias, float* __restrict__ C,
                     int M, int N, int K) {
  __shared__ float As[2][64 * APAD];
  __shared__ float Bs[2][32 * 64];

  const int tid  = threadIdx.x;
  const int wave = tid >> 5;
  const int lane = tid & 31;
  const int ln   = lane & 15;
  const int hi   = lane >> 4;
  const int nbN  = N >> 6;
  const int m0   = (blockIdx.x / nbN) * 64;
  const int n0   = (blockIdx.x % nbN) * 64;
  const int wm   = wave & 3;   // 4 m-waves
  const int wn   = wave >> 2;  // 2 n-waves

  // stage one 64x32 A panel + 32x64 B panel (4 async b128 per thread)
  auto issue = [&](int k0, int bsel) {
#pragma unroll
    for (int j = 0; j < 2; ++j) {  // A: 512 chunks of 16B
      const int id = j * 256 + tid;
      const int row = id >> 3, c4 = (id & 7) * 4;
      async_g2l_b128(&As[bsel][row * APAD + c4],
                     A + (size_t)(m0 + row) * K + k0 + c4);
    }
#pragma unroll
    for (int j = 0; j < 2; ++j) {  // B: 512 chunks of 16B
      const int id = j * 256 + tid;
      const int row = id >> 4, c4 = (id & 15) * 4;
      async_g2l_b128(&Bs[bsel][row * 64 + c4],
                     W + (size_t)(k0 + row) * N + n0 + c4);
    }
  };

  v8f acc0 = {}, acc1 = {};
  const int nch = K >> 5;
  issue(0, 0);
  wait_async<0>();
  __syncthreads();
  for (int c = 0; c < nch; ++c) {
    if (c + 1 < nch) issue((c + 1) * 32, (c + 1) & 1);  // overlap with compute
    const float* Ab = As[c & 1];
    const float* Bb = Bs[c & 1];
#pragma unroll
    for (int kk = 0; kk < 32; kk += 4) {
      const int ka = kk + 2 * hi;
      v2f a, b0, b1;
      a.x  = Ab[(wm * 16 + ln) * APAD + ka];
      a.y  = Ab[(wm * 16 + ln) * APAD + ka + 1];
      b0.x = Bb[ka * 64 + wn * 32 + ln];
      b0.y = Bb[(ka + 1) * 64 + wn * 32 + ln];
      b1.x = Bb[ka * 64 + wn * 32 + 16 + ln];
      b1.y = Bb[(ka + 1) * 64 + wn * 32 + 16 + ln];
      acc0 = __builtin_amdgcn_wmma_f32_16x16x4_f32(false, a, false, b0,
                                                   (short)0, acc0, false, false);
      acc1 = __builtin_amdgcn_wmma_f32_16x16x4_f32(false, a, false, b1,
                                                   (short)0, acc1, false, false);
    }
    wait_async<0>();   // own async loads for chunk c+1 done
    __syncthreads();   // all waves done computing c / loading c+1
  }

  const float bn0 = bias[n0 + wn * 32 + ln];
  const float bn1 = bias[n0 + wn * 32 + 16 + ln];
#pragma unroll
  for (int i = 0; i < 8; ++i) {
    const int row = m0 + wm * 16 + i + 8 * hi;
    C[(size_t)row * N + n0 + wn * 32 + ln]      = acc0[i] + bn0;
    C[(size_t)row * N + n0 + wn * 32 + 16 + ln] = acc1[i] + bn1;
  }
}

// ---------------------------------------------------------------------------
// Narrow GEMM for the syn projection (N=16): one 16x16 tile per wave.
// ---------------------------------------------------------------------------
__global__ __launch_bounds__(256)
void wmma_gemm_n16(const float* __restrict__ A, const float* __restrict__ W,
                   const float* __restrict__ bias, float* __restrict__ C,
                   int M, int N, int K) {
  const int wave = blockIdx.x * (blockDim.x >> 5) + (threadIdx.x >> 5);
  const int lane = threadIdx.x & 31;
  const int ln = lane & 15, hi = lane >> 4;
  const int ntN = N >> 4;
  const int mt = wave / ntN;
  const int nt = wave - mt * ntN;
  if (mt >= (M >> 4)) return;  // wave-uniform

  const float* arow = A + (size_t)(mt * 16 + ln) * K;
  const float* wcol = W + nt * 16 + ln;
  v8f acc = {};
  for (int k0 = 0; k0 < K; k0 += 4) {
    if ((k0 & 63) == 0 && k0 + 64 < K) __builtin_prefetch(arow + k0 + 64, 0, 0);
    const int ka = k0 + 2 * hi;
    v2f a, b;
    a.x = arow[ka];
    a.y = arow[ka + 1];
    b.x = wcol[(size_t)ka * N];
    b.y = wcol[(size_t)(ka + 1) * N];
    acc = __builtin_amdgcn_wmma_f32_16x16x4_f32(false, a, false, b, (short)0,
                                                acc, false, false);
  }
  const float bn = bias[nt * 16 + ln];
#pragma unroll
  for (int i = 0; i < 8; ++i)
    C[(size_t)(mt * 16 + i + 8 * hi) * N + nt * 16 + ln] = acc[i] + bn;
}

// ---------------------------------------------------------------------------
// Fused attention for one (batch, head, 16-row tile).
// Score stripe 16 x 2048 resident in LDS (rows padded to 2052 floats);
// K / V tiles staged per-wave through double-buffered async-to-LDS buffers.
// Weights hit HBM exactly once (the 536 MB store that bounds this op).
// ---------------------------------------------------------------------------
#define SP (SS + 4)  // score row stride (16B aligned, !=0 mod 64 banks)
#define QP 68        // Q/K tile row stride
__global__ __launch_bounds__(256)
void attn_wmma(const float* __restrict__ q, const float* __restrict__ k,
               const float* __restrict__ v, const float* __restrict__ syn,
               const int* __restrict__ mask, float* __restrict__ ctx,
               float* __restrict__ wts) {
  extern __shared__ float smem[];
  float* sc = smem;                 // [16][SP]
  float* qt = sc + 16 * SP;         // [16][QP]
  float* sy = qt + 16 * QP;         // [SS]
  float* cp = sy + SS;              // [2][16][HD]
  float* kt = cp + 2 * 16 * HD;     // 8 waves x 2 bufs x 16*QP (K/V staging)

  const int rt   = blockIdx.x & (SS / 16 - 1);
  const int h    = (blockIdx.x >> 7) & (HH - 1);
  const int b    = blockIdx.x >> 11;
  const int row0 = rt * 16;
  const int tid  = threadIdx.x;
  const int wave = tid >> 5;
  const int lane = tid & 31;
  const int ln   = lane & 15;
  const int hi   = lane >> 4;
  float* ktw = kt + wave * 2 * (16 * QP);

  for (int i = tid; i < 16 * HD; i += 256) {
    const int r = i >> 6, d = i & (HD - 1);
    qt[r * QP + d] = q[(size_t)(b * SS + row0 + r) * DD + h * HD + d];
  }
  for (int c = tid; c < SS; c += 256)
    sy[c] = syn[(size_t)(b * SS + c) * HH + h];
  __syncthreads();

  const float scale = 0.125f;  // 1/sqrt(64)
  const float* kb = k + (size_t)b * SS * DD + h * HD;

  // ---- scores: wave owns tiles ct = wave, wave+8, ...; K tile via async ---
  auto issue_k = [&](int ct, int bsel) {  // 16 rows x 64 floats, 8 b128/lane
    float* dst = ktw + bsel * (16 * QP);
#pragma unroll
    for (int j = 0; j < 8; ++j) {
      const int id = j * 32 + lane;
      const int row = id >> 4, c4 = (id & 15) * 4;
      async_g2l_b128(dst + row * QP + c4,
                     kb + (size_t)(ct * 16 + row) * DD + c4);
    }
  };
  issue_k(wave, 0);
  for (int ct = wave, it = 0; ct < SS / 16; ct += 8, ++it) {
    if (ct + 8 < SS / 16) { issue_k(ct + 8, (it + 1) & 1); wait_async<8>(); }
    else wait_async<0>();
    const float* kbuf = ktw + (it & 1) * (16 * QP);
    v8f acc = {};
#pragma unroll
    for (int kk = 0; kk < HD; kk += 4) {
      const int ka = kk + 2 * hi;
      v2f a, bm;
      a.x  = qt[ln * QP + ka];
      a.y  = qt[ln * QP + ka + 1];
      bm.x = kbuf[ln * QP + ka];
      bm.y = kbuf[ln * QP + ka + 1];
      acc = __builtin_amdgcn_wmma_f32_16x16x4_f32(false, a, false, bm, (short)0,
                                                  acc, false, false);
    }
    const int   c    = ct * 16 + ln;
    const float scol = sy[c];
    const float mneg = (mask[b * SS + c] == 0) ? -INFINITY : 0.0f;
#pragma unroll
    for (int i = 0; i < 8; ++i) {
      const int r = i + 8 * hi;
      sc[r * SP + c] = acc[i] * scale + sy[row0 + r] * scol + mneg;
    }
  }
  __syncthreads();

  // ---- prefetch first V tile (hides latency under softmax) ---------------
  const int nt = wave & 3;   // context N tile
  const int kh = wave >> 2;  // context K half
  const float* vb = v + (size_t)b * SS * DD + h * HD + nt * 16;
  auto issue_v = [&](int chunk, int bsel) {  // 32 rows x 16 floats, 4/lane
    float* dst = ktw + bsel * (16 * QP);
    const int k0 = kh * (SS / 2) + chunk * 32;
#pragma unroll
    for (int j = 0; j < 4; ++j) {
      const int id = j * 32 + lane;
      const int row = id >> 2, c4 = (id & 3) * 4;
      async_g2l_b128(dst + row * 16 + c4, vb + (size_t)(k0 + row) * DD + c4);
    }
  };
  issue_v(0, 0);

  // ---- softmax: wave w owns rows 2w, 2w+1 --------------------------------
  for (int rr = 0; rr < 2; ++rr) {
    const int r = wave * 2 + rr;
    float* srow = sc + r * SP;
    float mx = -INFINITY;
    for (int c = lane; c < SS; c += 32) mx = fmaxf(mx, srow[c]);
#pragma unroll
    for (int off = 16; off > 0; off >>= 1) mx = fmaxf(mx, __shfl_xor(mx, off, 32));
    float sum = 0.0f;
    for (int c = lane; c < SS; c += 32) {
      const float e = __expf(srow[c] - mx);
      srow[c] = e;
      sum += e;
    }
#pragma unroll
    for (int off = 16; off > 0; off >>= 1) sum += __shfl_xor(sum, off, 32);
    const float inv = 1.0f / sum;
    float* wrow = wts + ((size_t)(b * HH + h) * SS + row0 + r) * SS;
    for (int c = lane; c < SS; c += 32) {
      const float p = srow[c] * inv;
      srow[c] = p;   // kept for context GEMM
      wrow[c] = p;   // single HBM store of the weights output
    }
  }
  __syncthreads();

  // ---- context: ctx[16,64] = P[16,2048] @ V[2048,64] ----------------------
  // 8 waves = 4 N-tiles x 2 K-halves; V chunks double-buffered via async.
  {
    v8f acc = {};
    const int nch = (SS / 2) >> 5;  // 32 chunks of 32
    for (int c = 0; c < nch; ++c) {
      if (c + 1 < nch) { issue_v(c + 1, (c + 1) & 1); wait_async<4>(); }
      else wait_async<0>();
      const float* vbuf = ktw + (c & 1) * (16 * QP);
      const int kg0 = kh * (SS / 2) + c * 32;
#pragma unroll
      for (int kk = 0; kk < 32; kk += 4) {
        const int kl = kk + 2 * hi;
        v2f a, bm;
        a.x  = sc[ln * SP + kg0 + kl];
        a.y  = sc[ln * SP + kg0 + kl + 1];
        bm.x = vbuf[kl * 16 + ln];
        bm.y = vbuf[(kl + 1) * 16 + ln];
        acc = __builtin_amdgcn_wmma_f32_16x16x4_f32(false, a, false, bm,
                                                    (short)0, acc, false, false);
      }
    }
#pragma unroll
    for (int i = 0; i < 8; ++i)
      cp[(kh * 16 + i + 8 * hi) * HD + nt * 16 + ln] = acc[i];
  }
  __syncthreads();
  for (int i = tid; i < 16 * HD; i += 256) {
    const int r = i >> 6, d = i & (HD - 1);
    ctx[(size_t)(b * SS + row0 + r) * DD + h * HD + d] =
        cp[r * HD + d] + cp[(16 + r) * HD + d];
  }
}

// ---------------------------------------------------------------------------
extern "C" void kernel_launch(void* const* d_in, const int* in_sizes, int n_in,
                              void* d_out, int out_size, void* d_ws,
                              size_t ws_size, hipStream_t stream) {
  const float* query = (const float*)d_in[0];
  const float* key_  = (const float*)d_in[1];
  const float* value = (const float*)d_in[2];
  const int*   amask = (const int*)d_in[3];
  const float* syni  = (const float*)d_in[4];
  const float* Wq = (const float*)d_in[5];   const float* bq = (const float*)d_in[6];
  const float* Wk = (const float*)d_in[7];   const float* bk = (const float*)d_in[8];
  const float* Wv = (const float*)d_in[9];   const float* bv = (const float*)d_in[10];
  const float* Ws = (const float*)d_in[11];  const float* bs = (const float*)d_in[12];
  const float* Wo = (const float*)d_in[13];  const float* bo = (const float*)d_in[14];

  float* out = (float*)d_out;                  // [B,S,D]
  float* wts = out + (size_t)BB * SS * DD;     // [B,H,S,S]

  const size_t act = (size_t)BB * SS * DD;
  float* qws = (float*)d_ws;
  float* kws = qws + act;
  float* vws = kws + act;
  float* cws = vws + act;
  float* sws = cws + act;                      // [B*S, H]

  const int M = BB * SS;  // 4096
  const dim3 blk(256);

  const int blocksD = (M / 64) * (DD / 64);    // 1024 blocks
  wmma_gemm_tiled<<<blocksD, blk, 0, stream>>>(query, Wq, bq, qws, M, DD, DD);
  wmma_gemm_tiled<<<blocksD, blk, 0, stream>>>(key_,  Wk, bk, kws, M, DD, DD);
  wmma_gemm_tiled<<<blocksD, blk, 0, stream>>>(value, Wv, bv, vws, M, DD, DD);

  const int tilesH  = (M / 16) * (HH / 16);
  wmma_gemm_n16<<<(tilesH + 7) / 8, blk, 0, stream>>>(syni, Ws, bs, sws, M, HH, DD);

  const size_t ldsbytes =
      (size_t)(16 * SP + 16 * QP + SS + 2 * 16 * HD + 8 * 2 * 16 * QP) *
      sizeof(float);  // ~217 KB of the 320 KB WGP LDS
  attn_wmma<<<dim3(BB * HH * (SS / 16)), blk, ldsbytes, stream>>>(
      qws, kws, vws, sws, amask, cws, wts);

  wmma_gemm_tiled<<<blocksD, blk, 0, stream>>>(cws, Wo, bo, out, M, DD, DD);
}